// painn_45535243272754
// MI455X (gfx1250) — compile-verified
//
#include <hip/hip_runtime.h>
#include <hip/hip_bf16.h>

#define HID    128
#define PHI_N  384
#define NRBF   20
#define FCUT   5.0f
#define PI_F   3.14159265358979323846f

typedef __attribute__((ext_vector_type(2))) float v2f;
typedef __attribute__((ext_vector_type(8))) float v8f;

#if defined(__HIP_DEVICE_COMPILE__) && __has_builtin(__builtin_amdgcn_tensor_load_to_lds) && __has_builtin(__builtin_amdgcn_s_wait_tensorcnt)
#define USE_TDM 1
typedef unsigned int u32x4 __attribute__((ext_vector_type(4)));
typedef int          i32x4 __attribute__((ext_vector_type(4)));
typedef int          i32x8 __attribute__((ext_vector_type(8)));
#endif

// ---------------------------------------------------------------------------
// Kernel 1: node MLP  phi = silu(s @ W1 + b1) @ W2 + b2
// 4 waves / block, each wave owns one 16-row WMMA tile (64 rows per block)
// -> W1/W2 are re-read 256x instead of 1024x (~65 MB of L2 traffic).
// f32 WMMA (16x16x4) preserves reference precision; compute is free here.
// ---------------------------------------------------------------------------
__global__ __launch_bounds__(128) void painn_node_mlp(
    const float* __restrict__ s,
    const float* __restrict__ W1, const float* __restrict__ b1,
    const float* __restrict__ W2, const float* __restrict__ b2,
    float* __restrict__ phi)
{
    __shared__ float sA[4][16 * HID];   // 32 KB: input tiles, one per wave
    __shared__ float sH[4][16 * HID];   // 32 KB: hidden tiles, one per wave

    const int lane = threadIdx.x & 31;
    const int wv   = threadIdx.x >> 5;     // wave id 0..3
    const int m0   = blockIdx.x * 64 + wv * 16;
    const int hi   = lane >> 4;            // 0/1 half-wave
    const int mn   = lane & 15;            // row (A) / col (B,C,D)

    float* A = sA[wv];
    float* H = sH[wv];

    // coalesced stage of this wave's 16x128 input tile
    for (int i = lane; i < 16 * HID; i += 32) A[i] = s[(size_t)m0 * HID + i];
    __syncthreads();

    // ---- stage 1: h = silu(A @ W1 + b1), 8 column tiles of 16 ----
    for (int jn = 0; jn < 8; ++jn) {
        v8f acc = {};
        for (int k0 = 0; k0 < HID; k0 += 4) {
            v2f a, b;
            const int k = k0 + 2 * hi;
            a.x = A[mn * HID + k + 0];
            a.y = A[mn * HID + k + 1];
            b.x = W1[(k + 0) * HID + jn * 16 + mn];
            b.y = W1[(k + 1) * HID + jn * 16 + mn];
            acc = __builtin_amdgcn_wmma_f32_16x16x4_f32(
                false, a, false, b, (short)0, acc, false, false);
        }
#pragma unroll
        for (int r = 0; r < 8; ++r) {
            const int m = r + 8 * hi;
            const int n = jn * 16 + mn;
            const float x   = acc[r] + b1[n];
            const float sig = 1.0f / (1.0f + __expf(-x));
            H[m * HID + n] = x * sig;
        }
    }
    __syncthreads();

    // ---- stage 2: phi = H @ W2 + b2, 24 column tiles of 16 ----
    for (int jn = 0; jn < 24; ++jn) {
        v8f acc = {};
        for (int k0 = 0; k0 < HID; k0 += 4) {
            v2f a, b;
            const int k = k0 + 2 * hi;
            a.x = H[mn * HID + k + 0];
            a.y = H[mn * HID + k + 1];
            b.x = W2[(k + 0) * PHI_N + jn * 16 + mn];
            b.y = W2[(k + 1) * PHI_N + jn * 16 + mn];
            acc = __builtin_amdgcn_wmma_f32_16x16x4_f32(
                false, a, false, b, (short)0, acc, false, false);
        }
#pragma unroll
        for (int r = 0; r < 8; ++r) {
            const int m = r + 8 * hi;
            const int n = jn * 16 + mn;
            phi[(size_t)(m0 + m) * PHI_N + n] = acc[r] + b2[n];
        }
    }
}

// ---------------------------------------------------------------------------
// Kernel 2: edge stage (memory/atomic bound).
// Wrbf (30 KB) is DMA'd into LDS once per block via the Tensor Data Mover
// (wave 0 issues one tensor_load_to_lds descriptor, TENSORcnt-tracked),
// overlapping with brbf loads. 128 threads = 1 channel each; scatter via
// global_atomic_add_f32 (no-return -> STOREcnt path, L2-resident RMW).
// ---------------------------------------------------------------------------
__global__ __launch_bounds__(128) void painn_edge(
    const float* __restrict__ phi, const float* __restrict__ v,
    const float* __restrict__ r_ij,
    const int* __restrict__ esrc, const int* __restrict__ edst,
    const float* __restrict__ Wrbf, const float* __restrict__ brbf,
    float* __restrict__ dv, float* __restrict__ ds,
    int n_edges, int edges_per_block)
{
    __shared__ float sW[NRBF * PHI_N];   // 30 KB
    __shared__ float sB[PHI_N];          // 1.5 KB
    __shared__ float sRbf[NRBF];
    __shared__ float sGeo[4];            // unit xyz, fcut

    const int t = threadIdx.x;

#if defined(USE_TDM)
    if (t < 32) {   // wave 0 only issues the TDM descriptor
        // LDS aperture: low 32 bits of the generic address == LDS byte offset
        const unsigned lds_off = (unsigned)(size_t)(void*)&sW[0];
        const unsigned long long ga = (unsigned long long)(const void*)Wrbf;
        const unsigned nelem = (unsigned)(NRBF * PHI_N);   // 7680 f32

        u32x4 g0 = { 1u,                                   // count=1, user D#
                     lds_off,                              // lds_addr
                     (unsigned)ga,                         // global_addr[31:0]
                     (unsigned)((ga >> 32) & 0x1FFFFFFu) | 0x80000000u }; // addr[56:32] | type=2
        i32x8 g1 = { (int)(2u << 16),                      // data_size=4B
                     (int)((nelem & 0xFFFFu) << 16),       // tensor_dim0 lo16
                     (int)(1u << 16),                      // dim0 hi16=0 | tensor_dim1=1
                     (int)((nelem & 0xFFFFu) << 16),       // dim1 hi=0 | tile_dim0
                     1,                                    // tile_dim1=1, tile_dim2=0
                     (int)nelem,                           // dim0_stride[31:0]
                     0, 0 };
        i32x4 gz4 = { 0, 0, 0, 0 };
#if __clang_major__ >= 23
        i32x8 gz8 = { 0, 0, 0, 0, 0, 0, 0, 0 };
        __builtin_amdgcn_tensor_load_to_lds(g0, g1, gz4, gz4, gz8, 0);
#else
        __builtin_amdgcn_tensor_load_to_lds(g0, g1, gz4, gz4, 0);
#endif
        __builtin_amdgcn_s_wait_tensorcnt(0);
    }
    for (int i = t; i < PHI_N; i += 128) sB[i] = brbf[i];
#else
    for (int i = t; i < NRBF * PHI_N; i += 128) sW[i] = Wrbf[i];
    for (int i = t; i < PHI_N;        i += 128) sB[i] = brbf[i];
#endif

    const int e0 = blockIdx.x * edges_per_block;
    const int e1 = min(e0 + edges_per_block, n_edges);

    for (int e = e0; e < e1; ++e) {
        __syncthreads();                    // orders sW/sB staging + sRbf reuse
        if (t < NRBF) {
            const float rx = r_ij[(size_t)e * 3 + 0];
            const float ry = r_ij[(size_t)e * 3 + 1];
            const float rz = r_ij[(size_t)e * 3 + 2];
            const float d  = sqrtf(rx * rx + ry * ry + rz * rz);
            sRbf[t] = __sinf((float)(t + 1) * PI_F * d / FCUT) / d;
            if (t == 0) {
                const float inv = 1.0f / d;
                sGeo[0] = rx * inv; sGeo[1] = ry * inv; sGeo[2] = rz * inv;
                sGeo[3] = (d < FCUT) ? 0.5f * (__cosf(PI_F * d / FCUT) + 1.0f)
                                     : 0.0f;
            }
        }
        __syncthreads();

        // K=20 matvec against LDS-resident weights (3 output slices)
        float w_vv = sB[t], w_s = sB[HID + t], w_dir = sB[2 * HID + t];
#pragma unroll
        for (int n = 0; n < NRBF; ++n) {
            const float rn = sRbf[n];
            const float* row = &sW[n * PHI_N];
            w_vv  += rn * row[t];
            w_s   += rn * row[HID + t];
            w_dir += rn * row[2 * HID + t];
        }

        const float f   = sGeo[3];
        const int   src = esrc[e];
        const int   dst = edst[e];
        const float* pj = phi + (size_t)src * PHI_N;
        const float x_vv  = pj[t]           * (w_vv  * f);
        const float x_s   = pj[HID + t]     * (w_s   * f);
        const float x_dir = pj[2 * HID + t] * (w_dir * f);

        atomicAdd(&ds[(size_t)dst * HID + t], x_s);

        const float* vs  = v  + (size_t)src * 3 * HID;
        float*       dvd = dv + (size_t)dst * 3 * HID;
#pragma unroll
        for (int k = 0; k < 3; ++k) {
            atomicAdd(&dvd[k * HID + t],
                      vs[k * HID + t] * x_vv + sGeo[k] * x_dir);
        }
    }
}

// ---------------------------------------------------------------------------
extern "C" void kernel_launch(void* const* d_in, const int* in_sizes, int n_in,
                              void* d_out, int out_size, void* d_ws, size_t ws_size,
                              hipStream_t stream) {
    const float* s     = (const float*)d_in[0];
    const float* v     = (const float*)d_in[1];
    const float* r_ij  = (const float*)d_in[2];
    const int*   esrc  = (const int*)  d_in[3];
    const int*   edst  = (const int*)  d_in[4];
    const float* W1    = (const float*)d_in[5];
    const float* b1    = (const float*)d_in[6];
    const float* W2    = (const float*)d_in[7];
    const float* b2    = (const float*)d_in[8];
    const float* Wrbf  = (const float*)d_in[9];
    const float* brbf  = (const float*)d_in[10];

    const int n_nodes = in_sizes[0] / HID;
    const int n_edges = in_sizes[3];

    float* phi = (float*)d_ws;                       // 16384*384 f32 = 25 MB
    float* dv  = (float*)d_out;                      // (n_nodes,3,HID)
    float* ds  = dv + (size_t)n_nodes * 3 * HID;     // (n_nodes,HID)

    // outputs are accumulated with atomics -> zero them every call
    hipMemsetAsync(d_out, 0, (size_t)out_size * sizeof(float), stream);

    painn_node_mlp<<<n_nodes / 64, 128, 0, stream>>>(s, W1, b1, W2, b2, phi);

    const int nblocks = 8192;
    const int epb = (n_edges + nblocks - 1) / nblocks;
    painn_edge<<<nblocks, 128, 0, stream>>>(phi, v, r_ij, esrc, edst,
                                            Wrbf, brbf, dv, ds, n_edges, epb);
}